// ApplyCoeffs_23261542875222
// MI455X (gfx1250) — compile-verified
//
#include <hip/hip_runtime.h>
#include <stdint.h>

#define C_IN   4
#define C_OUT  31
#define CTOT   (C_IN * C_OUT)    // 124 coeff channels
#define HH     512
#define WW     512
#define HW     (HH * WW)         // 262144 pixels per image
#define NB     4                 // batch

#define BLOCK  128               // threads per block (4 waves)
#define WAVESZ 32
#define TILE_PIX 32              // pixels handled by one wave
#define OA     16                // outputs covered by descriptor A
#define OB     (C_OUT - OA)      // 15, covered by descriptor B

typedef __attribute__((ext_vector_type(4))) unsigned int u32x4;
typedef __attribute__((ext_vector_type(8))) int          i32x8;
typedef __attribute__((ext_vector_type(4))) int          i32x4;

// Build group1 for a 3D tile: X = 32 px (stride 1), Y = n_o output channels
// (stride HW), Z = 4 input channels (stride 31*HW). All dims in elements.
__device__ __forceinline__ i32x8 make_g1(int n_o)
{
    const long long s1 = (long long)C_OUT * HW;   // 8126464 = 0x7C0000
    i32x8 g1;
    g1[0] = (int)(2u << 16);                  // mask=0, data_size=2 (4B), no flags
    g1[1] = (int)((unsigned)TILE_PIX << 16);  // tensor_dim0[15:0] = 32
    g1[2] = (int)((unsigned)n_o << 16);       // tensor_dim0 hi = 0 ; tensor_dim1 lo = n_o
    g1[3] = (int)((unsigned)TILE_PIX << 16);  // tensor_dim1 hi = 0 ; tile_dim0 = 32
    g1[4] = (int)((4u << 16) | (unsigned)n_o);// tile_dim1 = n_o ; tile_dim2 = 4
    g1[5] = (int)HW;                          // tensor_dim0_stride[31:0]
    g1[6] = (int)(((unsigned)(s1 & 0xFFFF)) << 16);        // stride1[15:0] (=0)
    g1[7] = (int)(unsigned)(s1 >> 16);        // stride1[47:16] = 0x7C
    return g1;
}

__global__ __launch_bounds__(BLOCK)
void apply_coeffs_tdm(const float* __restrict__ coeff,
                      const float* __restrict__ inp,
                      float* __restrict__ out)
{
    // Per-wave slice: desc A fills [i][o 0..15][lane], desc B fills
    // [i][o 16..30][lane] right after it. 124*32*4 B per wave = 62 KB/block.
    __shared__ float smem[CTOT * BLOCK];

    const int tid  = threadIdx.x;
    const int lane = tid & (WAVESZ - 1);
    const int wave = tid >> 5;

    // Global pixel index; blocks never straddle an image (HW % BLOCK == 0).
    const long long pix0 = (long long)blockIdx.x * BLOCK + (long long)wave * TILE_PIX;
    const int n  = (int)(pix0 / HW);
    const int p0 = (int)(pix0 % HW);       // wave's first pixel within image n
    const int p  = p0 + lane;              // this lane's pixel

    const uint64_t gbase = (uint64_t)(uintptr_t)coeff +
        ((uint64_t)n * CTOT * HW + (uint64_t)p0) * 4ull;
    const uint32_t lbase = (uint32_t)(uintptr_t)smem
                         + (uint32_t)(wave * CTOT * TILE_PIX * 4);

    // ---- group0 helper values --------------------------------------------
    // A: channels i*31 + [0,16)  at gbase,                LDS lbase
    // B: channels i*31 + [16,31) at gbase + 16*HW*4,      LDS lbase + 4*16*32*4
    const uint64_t gA = gbase;
    const uint64_t gB = gbase + (uint64_t)OA * HW * 4ull;
    const uint32_t lA = lbase;
    const uint32_t lB = lbase + (uint32_t)(C_IN * OA * TILE_PIX * 4);

    u32x4 g0a, g0b;
    g0a[0] = 1u; g0a[1] = lA; g0a[2] = (uint32_t)gA;
    g0a[3] = ((uint32_t)(gA >> 32) & 0x01FFFFFFu) | (2u << 30);
    g0b[0] = 1u; g0b[1] = lB; g0b[2] = (uint32_t)gB;
    g0b[3] = ((uint32_t)(gB >> 32) & 0x01FFFFFFu) | (2u << 30);

    const i32x8 g1a = make_g1(OA);
    const i32x8 g1b = make_g1(OB);

    i32x4 g2; g2[0] = 4; g2[1] = 0; g2[2] = 0; g2[3] = 0;  // tensor_dim2 = 4 (Z)
    i32x4 g3; g3[0] = 0; g3[1] = (int)(1u << 16); g3[2] = 0; g3[3] = 0; // dim4=1 unused
    i32x8 gz;
    gz[0] = 0; gz[1] = 0; gz[2] = 0; gz[3] = 0;
    gz[4] = 0; gz[5] = 0; gz[6] = 0; gz[7] = 0;

    // Two TDM ops per wave, in flight simultaneously (in-order completion).
    __builtin_amdgcn_tensor_load_to_lds(g0a, g1a, g2, g3, gz, /*cpol=*/0);
    __builtin_amdgcn_tensor_load_to_lds(g0b, g1b, g2, g3, gz, /*cpol=*/0);

    // Read-once RGB planes while the DMAs run; stream past the caches.
    const float* ib = inp + (size_t)n * 3 * HW + (size_t)p;
    const float r = __builtin_nontemporal_load(ib + 0 * HW);
    const float g = __builtin_nontemporal_load(ib + 1 * HW);
    const float b = __builtin_nontemporal_load(ib + 2 * HW);

    const float* sA = &smem[wave * CTOT * TILE_PIX + lane];
    const float* sB = sA + C_IN * OA * TILE_PIX;
    float* obase = out + (size_t)n * C_OUT * HW + (size_t)p;

    // ---- phase 1: descriptor A landed (B may still be in flight) ---------
    __builtin_amdgcn_s_wait_tensorcnt(1);
    asm volatile("" ::: "memory");
#pragma unroll
    for (int o = 0; o < OA; ++o) {
        float acc = sA[(3 * OA + o) * TILE_PIX];             // homogeneous term
        acc = fmaf(r, sA[(0 * OA + o) * TILE_PIX], acc);
        acc = fmaf(g, sA[(1 * OA + o) * TILE_PIX], acc);
        acc = fmaf(b, sA[(2 * OA + o) * TILE_PIX], acc);
        __builtin_nontemporal_store(acc, obase + (size_t)o * HW);
    }

    // ---- phase 2: descriptor B landed ------------------------------------
    __builtin_amdgcn_s_wait_tensorcnt(0);
    asm volatile("" ::: "memory");
#pragma unroll
    for (int oo = 0; oo < OB; ++oo) {
        float acc = sB[(3 * OB + oo) * TILE_PIX];
        acc = fmaf(r, sB[(0 * OB + oo) * TILE_PIX], acc);
        acc = fmaf(g, sB[(1 * OB + oo) * TILE_PIX], acc);
        acc = fmaf(b, sB[(2 * OB + oo) * TILE_PIX], acc);
        __builtin_nontemporal_store(acc, obase + (size_t)(OA + oo) * HW);
    }
}

extern "C" void kernel_launch(void* const* d_in, const int* in_sizes, int n_in,
                              void* d_out, int out_size, void* d_ws, size_t ws_size,
                              hipStream_t stream)
{
    const float* coeff = (const float*)d_in[0];   // (4, 124, 512, 512) f32
    const float* inp   = (const float*)d_in[1];   // (4, 3, 512, 512)  f32
    float*       out   = (float*)d_out;           // (4, 31, 512, 512) f32

    const unsigned total_pix = NB * HW;           // 1,048,576
    dim3 grid(total_pix / BLOCK);                 // 8192 blocks
    dim3 block(BLOCK);
    apply_coeffs_tdm<<<grid, block, 0, stream>>>(coeff, inp, out);
}